// GNN_ogbproteins_4896262717705
// MI455X (gfx1250) — compile-verified
//
#include <hip/hip_runtime.h>
#include <hip/hip_bf16.h>

// ---------------------------------------------------------------------------
// GNN (SpGAT x2 + residual GC) for MI455X / gfx1250, fp32 WMMA matrix path.
// N=50000 nodes, E=800000 edges, NFEAT=8, NHID=128, NCLASS=112.
// All hot buffers fit in the 192MB L2; segment sums use global f32 atomics.
// ---------------------------------------------------------------------------

#define NN      50000
#define EE      800000
#define NFEAT   8
#define NHID    128
#define NCLASS  112

typedef __attribute__((ext_vector_type(2))) float v2f;
typedef __attribute__((ext_vector_type(8))) float v8f;

// ---------------------------------------------------------------------------
// Generic fp32 WMMA GEMM: C[M x NCOLS] = A[M x K] @ B[K x NCOLS].
// One wave -> one 16x16 C tile via V_WMMA_F32_16X16X4_F32 K-loop.
// blockDim.x = 32*(NCOLS/16); grid.x = M/16 (M multiple of 16).
// ISA VGPR layouts (CDNA5 ISA):
//   A 16x4 f32 : lane L -> M=L&15 ; vgpr0/1 = K {2*half, 2*half+1}, half=L>>4
//   B  4x16 f32: lane L -> N=L&15 ; vgpr0/1 = K {2*half, 2*half+1}
//   C/D 16x16  : vgpr r -> row r + 8*half ; col = L&15
// ---------------------------------------------------------------------------
template <int K, int NC>
__global__ __launch_bounds__(32 * (NC / 16))
void wmma_gemm_kernel(const float* __restrict__ A, const float* __restrict__ B,
                      float* __restrict__ C) {
    const int lane = threadIdx.x & 31;
    const int wave = threadIdx.x >> 5;
    const int m    = lane & 15;
    const int half = lane >> 4;
    const int row0 = blockIdx.x * 16;
    const int col0 = wave * 16;
    const float* __restrict__ arow = A + (size_t)(row0 + m) * K;

    v8f acc = {};
#pragma unroll
    for (int kb = 0; kb < K / 4; ++kb) {
        const int k0 = kb * 4 + 2 * half;
        v2f a; a.x = arow[k0];                  a.y = arow[k0 + 1];
        v2f b; b.x = B[k0 * NC + col0 + m];     b.y = B[(k0 + 1) * NC + col0 + m];
        acc = __builtin_amdgcn_wmma_f32_16x16x4_f32(false, a, false, b,
                                                    (short)0, acc, false, false);
    }
#pragma unroll
    for (int r = 0; r < 8; ++r)
        C[(size_t)(row0 + r + 8 * half) * NC + col0 + m] = acc[r];
}

// ---------------------------------------------------------------------------
// Output GEMM (K=128, NC=112) with fused residual epilogue:
//   support = H @ Wg ; out = support*(2/3) + bg   (SMOOTH=0.5)
// ---------------------------------------------------------------------------
__global__ __launch_bounds__(224)
void wmma_gemm_out_kernel(const float* __restrict__ A, const float* __restrict__ B,
                          const float* __restrict__ bg,
                          float* __restrict__ support, float* __restrict__ out) {
    const int lane = threadIdx.x & 31;
    const int wave = threadIdx.x >> 5;
    const int m    = lane & 15;
    const int half = lane >> 4;
    const int row0 = blockIdx.x * 16;
    const int col0 = wave * 16;
    const float* __restrict__ arow = A + (size_t)(row0 + m) * NHID;

    v8f acc = {};
#pragma unroll
    for (int kb = 0; kb < NHID / 4; ++kb) {
        const int k0 = kb * 4 + 2 * half;
        v2f a; a.x = arow[k0];                      a.y = arow[k0 + 1];
        v2f b; b.x = B[k0 * NCLASS + col0 + m];     b.y = B[(k0 + 1) * NCLASS + col0 + m];
        acc = __builtin_amdgcn_wmma_f32_16x16x4_f32(false, a, false, b,
                                                    (short)0, acc, false, false);
    }
    const float bias = bg[col0 + m];
#pragma unroll
    for (int r = 0; r < 8; ++r) {
        const size_t o = (size_t)(row0 + r + 8 * half) * NCLASS + col0 + m;
        const float v = acc[r];
        support[o] = v;
        out[o]     = v * (2.0f / 3.0f) + bias;   // support/(1+SMOOTH) + bias
    }
}

// ---------------------------------------------------------------------------
// Per-node attention pre-scores: fsrc[i] = h[i].a_src ; fdst[i] = h[i].a_dst
// ---------------------------------------------------------------------------
__global__ void fvec_kernel(const float* __restrict__ h,
                            const float* __restrict__ asrc, const float* __restrict__ adst,
                            float* __restrict__ fsrc, float* __restrict__ fdst) {
    const int i = blockIdx.x * blockDim.x + threadIdx.x;
    if (i >= NN) return;
    const float4* __restrict__ hr = (const float4*)(h + (size_t)i * NHID);
    const float4* __restrict__ a4 = (const float4*)asrc;
    const float4* __restrict__ b4 = (const float4*)adst;
    float s = 0.f, t = 0.f;
#pragma unroll
    for (int q = 0; q < NHID / 4; ++q) {
        const float4 hv = hr[q], av = a4[q], bv = b4[q];
        s += hv.x * av.x + hv.y * av.y + hv.z * av.z + hv.w * av.w;
        t += hv.x * bv.x + hv.y * bv.y + hv.z * bv.z + hv.w * bv.w;
    }
    fsrc[i] = s; fdst[i] = t;
}

// ---------------------------------------------------------------------------
// Edge attention scatter: one thread per (edge, 4-feature chunk).
//   val = exp(-leakyrelu(fsrc[s]+fdst[d], 0.2))
//   num[d][:] += val*h[s][:]  (f32 L2 atomics) ; denom[d] += val (chunk 0)
// ---------------------------------------------------------------------------
__global__ void edge_attn_kernel(const int* __restrict__ src, const int* __restrict__ dst,
                                 const float* __restrict__ fsrc, const float* __restrict__ fdst,
                                 const float* __restrict__ h,
                                 float* __restrict__ num, float* __restrict__ denom) {
    const long long idx = (long long)blockIdx.x * blockDim.x + threadIdx.x;
    if (idx >= (long long)EE * (NHID / 4)) return;
    const int e = (int)(idx >> 5);       // NHID/4 == 32 chunks per edge
    const int q = (int)(idx & 31);
    const int s = src[e], d = dst[e];
    const float z  = fsrc[s] + fdst[d];
    const float lr = z > 0.f ? z : 0.2f * z;
    const float val = __expf(-lr);
    const float4 hv = *(const float4*)(h + (size_t)s * NHID + q * 4);
    float* nb = num + (size_t)d * NHID + q * 4;
    atomicAdd(nb + 0, val * hv.x);
    atomicAdd(nb + 1, val * hv.y);
    atomicAdd(nb + 2, val * hv.z);
    atomicAdd(nb + 3, val * hv.w);
    if (q == 0) atomicAdd(denom + d, val);
}

// ---------------------------------------------------------------------------
// Softmax-normalize + per-column sum/sumsq for BatchNorm (block = 128 cols).
// ---------------------------------------------------------------------------
__global__ __launch_bounds__(NHID)
void attn_norm_stats_kernel(const float* __restrict__ num, const float* __restrict__ denom,
                            float* __restrict__ hout,
                            float* __restrict__ colsum, float* __restrict__ colsumsq,
                            int rows_per_block) {
    const int f  = threadIdx.x;
    const int r0 = blockIdx.x * rows_per_block;
    int r1 = r0 + rows_per_block; if (r1 > NN) r1 = NN;
    float s = 0.f, ss = 0.f;
    for (int r = r0; r < r1; ++r) {
        const float v = num[(size_t)r * NHID + f] / (denom[r] + 1e-16f);
        hout[(size_t)r * NHID + f] = v;
        s += v; ss += v * v;
    }
    atomicAdd(colsum + f, s);
    atomicAdd(colsumsq + f, ss);
}

// ---------------------------------------------------------------------------
// BatchNorm apply + leaky_relu(0.01):  y = (x-mu)*rsqrt(var+1e-5)*gamma+beta
// ---------------------------------------------------------------------------
__global__ void bn_apply_kernel(const float* __restrict__ hin,
                                const float* __restrict__ colsum, const float* __restrict__ colsumsq,
                                const float* __restrict__ gamma, const float* __restrict__ beta,
                                float* __restrict__ hout) {
    const int idx = blockIdx.x * blockDim.x + threadIdx.x;
    if (idx >= NN * NHID) return;
    const int f = idx & (NHID - 1);
    const float invN = 1.0f / (float)NN;
    const float mu  = colsum[f] * invN;
    const float var = colsumsq[f] * invN - mu * mu;
    const float y = (hin[idx] - mu) * rsqrtf(var + 1e-5f) * gamma[f] + beta[f];
    hout[idx] = y > 0.f ? y : 0.01f * y;
}

// ---------------------------------------------------------------------------
// Residual graph conv scatter: out[d][:] += (ew[e]/3) * support[s][:]
// One thread per (edge, 4-class chunk); NCLASS/4 = 28 chunks.
// ---------------------------------------------------------------------------
__global__ void residual_edge_kernel(const int* __restrict__ src, const int* __restrict__ dst,
                                     const float* __restrict__ ew,
                                     const float* __restrict__ support, float* __restrict__ out) {
    const long long idx = (long long)blockIdx.x * blockDim.x + threadIdx.x;
    if (idx >= (long long)EE * (NCLASS / 4)) return;
    const int e = (int)(idx / (NCLASS / 4));
    const int q = (int)(idx - (long long)e * (NCLASS / 4));
    const int s = src[e], d = dst[e];
    const float w = ew[e] * (1.0f / 3.0f);   // SMOOTH/(1+SMOOTH)
    const float4 sv = *(const float4*)(support + (size_t)s * NCLASS + q * 4);
    float* ob = out + (size_t)d * NCLASS + q * 4;
    atomicAdd(ob + 0, w * sv.x);
    atomicAdd(ob + 1, w * sv.y);
    atomicAdd(ob + 2, w * sv.z);
    atomicAdd(ob + 3, w * sv.w);
}

// ---------------------------------------------------------------------------
extern "C" void kernel_launch(void* const* d_in, const int* in_sizes, int n_in,
                              void* d_out, int out_size, void* d_ws, size_t ws_size,
                              hipStream_t stream) {
    (void)in_sizes; (void)n_in; (void)out_size; (void)ws_size;
    const float* x     = (const float*)d_in[0];
    const int*   ei    = (const int*)d_in[1];
    const float* ew    = (const float*)d_in[2];
    const float* W1    = (const float*)d_in[3];
    const float* a1s   = (const float*)d_in[4];
    const float* a1d   = (const float*)d_in[5];
    const float* W2    = (const float*)d_in[6];
    const float* a2s   = (const float*)d_in[7];
    const float* a2d   = (const float*)d_in[8];
    const float* gamma = (const float*)d_in[9];
    const float* beta  = (const float*)d_in[10];
    const float* Wg    = (const float*)d_in[11];
    const float* bg    = (const float*)d_in[12];
    float* out = (float*)d_out;
    const int* src = ei;
    const int* dst = ei + EE;

    // workspace carve (floats): hA | hB | fsrc | fdst | denom | colsum | colsumsq | support
    float* hA = (float*)d_ws;
    float* hB = hA + (size_t)NN * NHID;
    float* fs = hB + (size_t)NN * NHID;
    float* fd = fs + NN;
    float* dn = fd + NN;
    float* cs = dn + NN;
    float* cq = cs + NHID;
    float* sp = cq + NHID;

    const int rowsTiles   = NN / 16;                       // 3125
    const int edgeThreads = EE * (NHID / 4);               // 25.6M
    const int resThreads  = EE * (NCLASS / 4);             // 22.4M

    // ---- attention layer 1 ----
    wmma_gemm_kernel<NFEAT, NHID><<<rowsTiles, 256, 0, stream>>>(x, W1, hA);
    fvec_kernel<<<(NN + 255) / 256, 256, 0, stream>>>(hA, a1s, a1d, fs, fd);
    hipMemsetAsync(hB, 0, (size_t)NN * NHID * sizeof(float), stream);
    hipMemsetAsync(dn, 0, (size_t)NN * sizeof(float), stream);
    edge_attn_kernel<<<(edgeThreads + 255) / 256, 256, 0, stream>>>(src, dst, fs, fd, hA, hB, dn);
    hipMemsetAsync(cs, 0, NHID * sizeof(float), stream);
    hipMemsetAsync(cq, 0, NHID * sizeof(float), stream);
    attn_norm_stats_kernel<<<1000, NHID, 0, stream>>>(hB, dn, hA, cs, cq, NN / 1000);
    bn_apply_kernel<<<(NN * NHID + 255) / 256, 256, 0, stream>>>(hA, cs, cq, gamma, beta, hB);

    // ---- attention layer 2 ----
    wmma_gemm_kernel<NHID, NHID><<<rowsTiles, 256, 0, stream>>>(hB, W2, hA);
    fvec_kernel<<<(NN + 255) / 256, 256, 0, stream>>>(hA, a2s, a2d, fs, fd);
    hipMemsetAsync(hB, 0, (size_t)NN * NHID * sizeof(float), stream);
    hipMemsetAsync(dn, 0, (size_t)NN * sizeof(float), stream);
    edge_attn_kernel<<<(edgeThreads + 255) / 256, 256, 0, stream>>>(src, dst, fs, fd, hA, hB, dn);
    hipMemsetAsync(cs, 0, NHID * sizeof(float), stream);
    hipMemsetAsync(cq, 0, NHID * sizeof(float), stream);
    attn_norm_stats_kernel<<<1000, NHID, 0, stream>>>(hB, dn, hA, cs, cq, NN / 1000);
    bn_apply_kernel<<<(NN * NHID + 255) / 256, 256, 0, stream>>>(hA, cs, cq, gamma, beta, hB);

    // ---- residual graph conv output ----
    wmma_gemm_out_kernel<<<rowsTiles, 224, 0, stream>>>(hB, Wg, bg, sp, out);
    residual_edge_kernel<<<(resThreads + 255) / 256, 256, 0, stream>>>(src, dst, ew, sp, out);
}